// Net_3642132267516
// MI455X (gfx1250) — compile-verified
//
#include <hip/hip_runtime.h>
#include <math.h>

// Problem constants (from reference)
#define NN 7650          // nodes
#define FF 745           // features
#define EE 238162        // edges
#define NP 7664          // N padded to 16  (479*16)
#define FP 752           // F padded to 16  (47*16)
#define NFPAD ((size_t)NP * (size_t)FP)

typedef __attribute__((ext_vector_type(2))) float v2f;
typedef __attribute__((ext_vector_type(8))) float v8f;

#define CDIV(a, b) (((a) + (b) - 1) / (b))

// ---------------------------------------------------------------- utilities
__global__ void k_zero(float* p, size_t n) {
  size_t i = (size_t)blockIdx.x * blockDim.x + threadIdx.x;
  if (i < n) p[i] = 0.f;
}

__global__ void k_ones(float* p, int n) {
  int i = blockIdx.x * blockDim.x + threadIdx.x;
  if (i < n) p[i] = 1.f;
}

// Pack X (NN x FF) -> Xp (NP x FP), zero padded
__global__ void k_packX(const float* __restrict__ X, float* __restrict__ Xp) {
  size_t i = (size_t)blockIdx.x * blockDim.x + threadIdx.x;
  if (i >= NFPAD) return;
  const int n = (int)(i / FP), f = (int)(i % FP);
  Xp[i] = (n < NN && f < FF) ? X[(size_t)n * FF + f] : 0.f;
}

// Pack W (FF x FF, row-major k,n) -> Wt (FP x FP, Wt[n][k] = W[k][n]), zero padded
__global__ void k_packWt(const float* __restrict__ W, float* __restrict__ Wt) {
  size_t i = (size_t)blockIdx.x * blockDim.x + threadIdx.x;
  if (i >= (size_t)FP * FP) return;
  const int n = (int)(i / FP), k = (int)(i % FP);
  Wt[i] = (n < FF && k < FF) ? W[(size_t)k * FF + n] : 0.f;
}

// ------------------------------------------------ degree + GCN edge weights
__global__ void k_deg(const int* __restrict__ row, const float* __restrict__ attr,
                      float* __restrict__ deg) {
  int e = blockIdx.x * blockDim.x + threadIdx.x;
  if (e < EE) atomicAdd(&deg[row[e]], attr[e]);
}

__device__ __forceinline__ float d_invsqrt(float d) {
  return d > 0.f ? 1.f / sqrtf(fmaxf(d, 1e-12f)) : 0.f;
}

__global__ void k_edgew(const int* __restrict__ row, const int* __restrict__ col,
                        const float* __restrict__ attr, const float* __restrict__ deg,
                        float* __restrict__ w) {
  int e = blockIdx.x * blockDim.x + threadIdx.x;
  if (e < EE) w[e] = d_invsqrt(deg[row[e]]) * attr[e] * d_invsqrt(deg[col[e]]);
}

// --------------------------------------------- Jacobi recurrence coef table
// coefs[L*4 + {0,1,2}] = {p_Ax, p_xm1, p_xm2}:  x_L = p0*Ax + p1*x_{L-1} + p2*x_{L-2}
__global__ void k_coefs(const float* __restrict__ alphas_raw, float* __restrict__ coefs) {
  if (threadIdx.x != 0 || blockIdx.x != 0) return;
  float al[4];
  for (int j = 0; j < 4; ++j) al[j] = tanhf(alphas_raw[j]);
  const float a = 1.f, b = 1.f, l = -1.f, r = 1.f;
  // L = 1
  float c1 = (a - b) * 0.5f - (a + b + 2.f) * 0.5f * (l + r) / (r - l);
  float c2 = (a + b + 2.f) / (r - l);
  coefs[4 + 0] = al[0] * c2;
  coefs[4 + 1] = al[0] * c1;
  coefs[4 + 2] = 0.f;
  // L = 2,3
  for (int L = 2; L <= 3; ++L) {
    float Lf = (float)L;
    float coef_l     = 2.f * Lf * (Lf + a + b) * (2.f * Lf - 2.f + a + b);
    float coef_lm1_1 = (2.f * Lf + a + b - 1.f) * (2.f * Lf + a + b) * (2.f * Lf + a + b - 2.f);
    float coef_lm1_2 = (2.f * Lf + a + b - 1.f) * (a * a - b * b);
    float coef_lm2   = 2.f * (Lf - 1.f + a) * (Lf - 1.f + b) * (2.f * Lf + a + b);
    float tmp1  = al[L - 1] * (coef_lm1_1 / coef_l);
    float tmp2  = al[L - 1] * (coef_lm1_2 / coef_l);
    float tmp3  = al[L - 1] * al[L - 2] * (coef_lm2 / coef_l);
    float tmp12 = tmp1 * (2.f / (r - l));
    float tmp22 = tmp1 * ((r + l) / (r - l)) + tmp2;
    coefs[L * 4 + 0] = tmp12;
    coefs[L * 4 + 1] = -tmp22;
    coefs[L * 4 + 2] = -tmp3;
  }
}

// --------------------------------------------------- GEMM: Y = Xp@W + bias (fp32 WMMA)
// Xp: NP x FP padded, Wt: FP x FP with Wt[n][k] = W[k][n] (both zero padded).
// Block = 256 thr = 8 waves stacked in M; wave owns 16(M) x 64(N), 4 WMMA accums.
// Software-pipelined: iteration i+1's 5 b64 loads are issued before iteration
// i's 4 WMMAs, so each s_wait at a WMMA covers loads launched one iter earlier.
__global__ void __launch_bounds__(256)
k_gemm_wmma(const float* __restrict__ Xp, const float* __restrict__ Wt,
            const float* __restrict__ bias, float* __restrict__ Y) {
  const int lane = threadIdx.x & 31;
  const int wave = threadIdx.x >> 5;
  const int rowl = lane & 15;       // M (A) / N (B,C) within tile
  const int hi   = lane >> 4;       // K-half selector
  const int m0 = blockIdx.y * 128 + wave * 16;
  const int n0 = blockIdx.x * 64;

  // Clamp load rows: tiles fully out of range produce garbage that the guarded
  // store discards; in-range tiles have no clamped lanes.
  const float* Arow  = Xp + (size_t)min(m0 + rowl, NP - 1) * FP + hi * 2;
  const float* Brow0 = Wt + (size_t)min(n0 +  0 + rowl, FP - 1) * FP + hi * 2;
  const float* Brow1 = Wt + (size_t)min(n0 + 16 + rowl, FP - 1) * FP + hi * 2;
  const float* Brow2 = Wt + (size_t)min(n0 + 32 + rowl, FP - 1) * FP + hi * 2;
  const float* Brow3 = Wt + (size_t)min(n0 + 48 + rowl, FP - 1) * FP + hi * 2;

  v8f acc0 = (v8f){}, acc1 = (v8f){}, acc2 = (v8f){}, acc3 = (v8f){};

  // Prologue: fragments for kk = 0
  v2f a_c  = *(const v2f*)(Arow);
  v2f b0_c = *(const v2f*)(Brow0);
  v2f b1_c = *(const v2f*)(Brow1);
  v2f b2_c = *(const v2f*)(Brow2);
  v2f b3_c = *(const v2f*)(Brow3);

  constexpr int NK = FP / 4;  // 188 k-steps
#pragma unroll 4
  for (int it = 0; it < NK - 1; ++it) {
    const int kk = (it + 1) * 4;
    // issue next-iteration loads first
    const v2f a_n  = *(const v2f*)(Arow + kk);
    const v2f b0_n = *(const v2f*)(Brow0 + kk);
    const v2f b1_n = *(const v2f*)(Brow1 + kk);
    const v2f b2_n = *(const v2f*)(Brow2 + kk);
    const v2f b3_n = *(const v2f*)(Brow3 + kk);
#if __has_builtin(__builtin_amdgcn_wmma_f32_16x16x4_f32)
    acc0 = __builtin_amdgcn_wmma_f32_16x16x4_f32(false, a_c, false, b0_c, (short)0, acc0, false, false);
    acc1 = __builtin_amdgcn_wmma_f32_16x16x4_f32(false, a_c, false, b1_c, (short)0, acc1, false, false);
    acc2 = __builtin_amdgcn_wmma_f32_16x16x4_f32(false, a_c, false, b2_c, (short)0, acc2, false, false);
    acc3 = __builtin_amdgcn_wmma_f32_16x16x4_f32(false, a_c, false, b3_c, (short)0, acc3, false, false);
#else
    acc0[0] += a_c.x * b0_c.x; acc1[0] += a_c.x * b1_c.x;  // compiling fallback only
    acc2[0] += a_c.x * b2_c.x; acc3[0] += a_c.x * b3_c.x;
#endif
    a_c = a_n; b0_c = b0_n; b1_c = b1_n; b2_c = b2_n; b3_c = b3_n;
  }
  // Epilogue: last k-step
#if __has_builtin(__builtin_amdgcn_wmma_f32_16x16x4_f32)
  acc0 = __builtin_amdgcn_wmma_f32_16x16x4_f32(false, a_c, false, b0_c, (short)0, acc0, false, false);
  acc1 = __builtin_amdgcn_wmma_f32_16x16x4_f32(false, a_c, false, b1_c, (short)0, acc1, false, false);
  acc2 = __builtin_amdgcn_wmma_f32_16x16x4_f32(false, a_c, false, b2_c, (short)0, acc2, false, false);
  acc3 = __builtin_amdgcn_wmma_f32_16x16x4_f32(false, a_c, false, b3_c, (short)0, acc3, false, false);
#else
  acc0[0] += a_c.x * b0_c.x; acc1[0] += a_c.x * b1_c.x;
  acc2[0] += a_c.x * b2_c.x; acc3[0] += a_c.x * b3_c.x;
#endif

  // C/D layout: VGPR i -> M = i (lanes 0-15) / M = 8+i (lanes 16-31), N = lane&15
#define STORE_TILE(ACC, T)                                                   \
  {                                                                          \
    const int colN = n0 + (T) * 16 + rowl;                                   \
    const float bia = (colN < FF) ? bias[colN] : 0.f;                        \
    _Pragma("unroll")                                                        \
    for (int i = 0; i < 8; ++i) {                                            \
      const int rowM = m0 + i + hi * 8;                                      \
      if (rowM < NP && colN < FP) Y[(size_t)rowM * FP + colN] = (ACC)[i] + bia; \
    }                                                                        \
  }
  STORE_TILE(acc0, 0)
  STORE_TILE(acc1, 1)
  STORE_TILE(acc2, 2)
  STORE_TILE(acc3, 3)
#undef STORE_TILE
}

// ------------------------------------------------------ SPMM (dense features)
// y[row[e], :] += w[e] * x[col[e], :]  over padded FP rows; float4 slices.
__global__ void k_spmm4(const int* __restrict__ row, const int* __restrict__ col,
                        const float* __restrict__ w, const float* __restrict__ x,
                        float* __restrict__ y) {
  constexpr int CH = FP / 4;  // 188 float4 chunks per row
  long long idx = (long long)blockIdx.x * blockDim.x + threadIdx.x;
  if (idx >= (long long)EE * CH) return;
  const int e = (int)(idx / CH);
  const int c = (int)(idx % CH);
  const float we = w[e];
  const float4 v = *(const float4*)(x + (size_t)col[e] * FP + c * 4);
  float* ys = y + (size_t)row[e] * FP + c * 4;
  atomicAdd(ys + 0, we * v.x);
  atomicAdd(ys + 1, we * v.y);
  atomicAdd(ys + 2, we * v.z);
  atomicAdd(ys + 3, we * v.w);
}

// scalar SPMM for the rank-1 shortcut: g-vector propagation
__global__ void k_spmm_s(const int* __restrict__ row, const int* __restrict__ col,
                         const float* __restrict__ w, const float* __restrict__ x,
                         float* __restrict__ y) {
  int e = blockIdx.x * blockDim.x + threadIdx.x;
  if (e < EE) atomicAdd(&y[row[e]], w[e] * x[col[e]]);
}

// ----------------------------------------------- Jacobi step + accumulation
// xnew = p0*Ax + p1*xm1 + p2*xm2 ; dst = xnew ; out = init ? xm1+xnew : out+xnew
__global__ void k_jacobi(const float* __restrict__ coef3,
                         const float* __restrict__ xm1, const float* __restrict__ xm2,
                         const float* __restrict__ Ax, float* __restrict__ dst,
                         float* __restrict__ out, int init, size_t count) {
  size_t i = (size_t)blockIdx.x * blockDim.x + threadIdx.x;
  if (i >= count) return;
  const float p0 = coef3[0], p1 = coef3[1], p2 = coef3[2];
  const float xnew = p0 * Ax[i] + p1 * xm1[i] + p2 * xm2[i];
  dst[i] = xnew;
  out[i] = init ? (xm1[i] + xnew) : (out[i] + xnew);
}

// ----------------------------------------------------------- Tucker head
// P0[r,f] = sum_n U0[n,r] * t0[n,f]   (one block per feature column)
__global__ void __launch_bounds__(256)
k_headP0(const float* __restrict__ t0, const float* __restrict__ U0,
         float* __restrict__ P0) {
  const int f = blockIdx.x;  // 0..FF-1
  float acc[8];
#pragma unroll
  for (int r = 0; r < 8; ++r) acc[r] = 0.f;
  for (int n = threadIdx.x; n < NN; n += 256) {
    const float x = t0[(size_t)n * FP + f];
#pragma unroll
    for (int r = 0; r < 8; ++r) acc[r] += U0[n * 8 + r] * x;
  }
  __shared__ float red[256];
  for (int r = 0; r < 8; ++r) {
    red[threadIdx.x] = acc[r];
    __syncthreads();
    for (int s = 128; s > 0; s >>= 1) {
      if (threadIdx.x < s) red[threadIdx.x] += red[threadIdx.x + s];
      __syncthreads();
    }
    if (threadIdx.x == 0) P0[r * FF + f] = red[0];
    __syncthreads();
  }
}

// h[r] = sum_n U0[n,r] * g[n]
__global__ void __launch_bounds__(256)
k_headH(const float* __restrict__ g, const float* __restrict__ U0, float* __restrict__ h) {
  float acc[8];
#pragma unroll
  for (int r = 0; r < 8; ++r) acc[r] = 0.f;
  for (int n = threadIdx.x; n < NN; n += 256) {
    const float x = g[n];
#pragma unroll
    for (int r = 0; r < 8; ++r) acc[r] += U0[n * 8 + r] * x;
  }
  __shared__ float red[256];
  for (int r = 0; r < 8; ++r) {
    red[threadIdx.x] = acc[r];
    __syncthreads();
    for (int s = 128; s > 0; s >>= 1) {
      if (threadIdx.x < s) red[threadIdx.x] += red[threadIdx.x + s];
      __syncthreads();
    }
    if (threadIdx.x == 0) h[r] = red[0];
    __syncthreads();
  }
}

// Final tiny contractions + log_softmax -> d_out[8]
__global__ void __launch_bounds__(64)
k_final(const float* __restrict__ P0, const float* __restrict__ h,
        const float* __restrict__ b2, const float* __restrict__ b3,
        const float* __restrict__ U1, const float* __restrict__ U2,
        const float* __restrict__ core, const float* __restrict__ U3,
        const float* __restrict__ trl_bias, float* __restrict__ out) {
  __shared__ float sQ0[24], sB2[3], sB3[3];
  const int tid = threadIdx.x;
  if (tid < 24) {                 // Q0[r,s] = sum_f P0[r,f] U1[f,s]
    const int r = tid / 3, s = tid % 3;
    float q = 0.f;
    for (int f = 0; f < FF; ++f) q += P0[r * FF + f] * U1[f * 3 + s];
    sQ0[tid] = q;
  } else if (tid < 27) {          // B2s[s] = sum_f b2[f] U1[f,s]
    const int s = tid - 24;
    float q = 0.f;
    for (int f = 0; f < FF; ++f) q += b2[f] * U1[f * 3 + s];
    sB2[s] = q;
  } else if (tid < 30) {
    const int s = tid - 27;
    float q = 0.f;
    for (int f = 0; f < FF; ++f) q += b3[f] * U1[f * 3 + s];
    sB3[s] = q;
  }
  __syncthreads();
  if (tid == 0) {
    float v[8];
#pragma unroll
    for (int u = 0; u < 8; ++u) v[u] = 0.f;
    for (int r = 0; r < 8; ++r)
      for (int s = 0; s < 3; ++s)
        for (int t = 0; t < 3; ++t) {
          // R[r,s,t] = sum_c Q[r,s,c]*U2[c,t]; Q[.,1]=h[r]*B2s, Q[.,2]=h[r]*B3s
          const float R = sQ0[r * 3 + s] * U2[0 * 3 + t] +
                          h[r] * sB2[s] * U2[1 * 3 + t] +
                          h[r] * sB3[s] * U2[2 * 3 + t];
          for (int u = 0; u < 8; ++u) v[u] += R * core[((r * 3 + s) * 3 + t) * 8 + u];
        }
    float o[8];
    float mx = -1e30f;
    for (int oo = 0; oo < 8; ++oo) {
      float s = trl_bias[0];
      for (int u = 0; u < 8; ++u) s += v[u] * U3[oo * 8 + u];
      o[oo] = s;
      mx = fmaxf(mx, s);
    }
    float lse = 0.f;
    for (int oo = 0; oo < 8; ++oo) lse += expf(o[oo] - mx);
    lse = logf(lse);
    for (int oo = 0; oo < 8; ++oo) out[oo] = o[oo] - mx - lse;
  }
}

// ================================================================= launcher
extern "C" void kernel_launch(void* const* d_in, const int* in_sizes, int n_in,
                              void* d_out, int out_size, void* d_ws, size_t ws_size,
                              hipStream_t stream) {
  // inputs (setup_inputs order)
  const float* X     = (const float*)d_in[0];
  const int*   eidx  = (const int*)d_in[1];      // (2,E): row then col
  const float* eattr = (const float*)d_in[2];
  const float* W1    = (const float*)d_in[3];
  const float* b1    = (const float*)d_in[4];
  // d_in[5]=W2, d_in[7]=W3: mathematically unused (zeros @ W == 0)
  const float* b2    = (const float*)d_in[6];
  const float* b3    = (const float*)d_in[8];
  const float* alr   = (const float*)d_in[9];
  const float* core  = (const float*)d_in[10];
  const float* U0    = (const float*)d_in[11];
  const float* U1    = (const float*)d_in[12];
  const float* U2    = (const float*)d_in[13];
  const float* U3    = (const float*)d_in[14];
  const float* trlb  = (const float*)d_in[15];
  float* outp = (float*)d_out;

  const int* erow = eidx;
  const int* ecol = eidx + EE;

  // workspace carve-up (floats); total ~96 MB (fits in L2: 192 MB)
  float* f   = (float*)d_ws;
  float* bufA  = f;                 // x_{L-1} / x0 (padded NP x FP)
  float* bufB  = bufA + NFPAD;      // x_{L-2} / x_new
  float* bufO  = bufB + NFPAD;      // running sum (t0)
  float* bufAx = bufO + NFPAD;      // Xp during GEMM, then SPMM target
  float* deg   = bufAx + NFPAD;     // NN
  float* wEd   = deg + NN;          // EE
  float* g0    = wEd + EE;          // NN (scalar polyconv buffers)
  float* g1    = g0 + NN;
  float* gO    = g1 + NN;
  float* gAx   = gO + NN;
  float* coefs = gAx + NN;          // 16
  float* P0    = coefs + 16;        // 8 * FF
  float* hbuf  = P0 + 8 * FF;       // 8
  float* Wt    = hbuf + 8;          // FP * FP (transposed, padded W1)

  const int T = 256;
  const size_t spmmT = (size_t)EE * (FP / 4);

  // 1) normalized edge weights + recurrence coefs
  k_zero<<<CDIV(NN, T), T, 0, stream>>>(deg, (size_t)NN);
  k_deg<<<CDIV(EE, T), T, 0, stream>>>(erow, eattr, deg);
  k_edgew<<<CDIV(EE, T), T, 0, stream>>>(erow, ecol, eattr, deg, wEd);
  k_coefs<<<1, 1, 0, stream>>>(alr, coefs);

  // 2) pack operands; x0 = X @ W1 + b1  (fp32 WMMA GEMM, pipelined inner loop)
  k_packX<<<CDIV(NFPAD, T), T, 0, stream>>>(X, bufAx);            // Xp in bufAx
  k_packWt<<<CDIV((size_t)FP * FP, (size_t)T), T, 0, stream>>>(W1, Wt);
  {
    dim3 grid(CDIV(FP, 64), CDIV(NP, 128));
    k_gemm_wmma<<<grid, 256, 0, stream>>>(bufAx, Wt, b1, bufA);
  }

  // 3) dense poly conv: t0 = sum_L x_L   (buffers rotate; dst aliases xm2)
  // L=1
  k_zero<<<CDIV(NFPAD, T), T, 0, stream>>>(bufAx, NFPAD);
  k_spmm4<<<CDIV(spmmT, (size_t)T), T, 0, stream>>>(erow, ecol, wEd, bufA, bufAx);
  k_jacobi<<<CDIV(NFPAD, T), T, 0, stream>>>(coefs + 4, bufA, bufA, bufAx, bufB, bufO, 1, NFPAD);
  // L=2
  k_zero<<<CDIV(NFPAD, T), T, 0, stream>>>(bufAx, NFPAD);
  k_spmm4<<<CDIV(spmmT, (size_t)T), T, 0, stream>>>(erow, ecol, wEd, bufB, bufAx);
  k_jacobi<<<CDIV(NFPAD, T), T, 0, stream>>>(coefs + 8, bufB, bufA, bufAx, bufA, bufO, 0, NFPAD);
  // L=3
  k_zero<<<CDIV(NFPAD, T), T, 0, stream>>>(bufAx, NFPAD);
  k_spmm4<<<CDIV(spmmT, (size_t)T), T, 0, stream>>>(erow, ecol, wEd, bufA, bufAx);
  k_jacobi<<<CDIV(NFPAD, T), T, 0, stream>>>(coefs + 12, bufA, bufB, bufAx, bufB, bufO, 0, NFPAD);
  // bufO == t0

  // 4) scalar poly conv for the rank-1 branches: g = polyconv(ones)
  k_ones<<<CDIV(NN, T), T, 0, stream>>>(g0, NN);
  k_zero<<<CDIV(NN, T), T, 0, stream>>>(gAx, (size_t)NN);
  k_spmm_s<<<CDIV(EE, T), T, 0, stream>>>(erow, ecol, wEd, g0, gAx);
  k_jacobi<<<CDIV(NN, T), T, 0, stream>>>(coefs + 4, g0, g0, gAx, g1, gO, 1, (size_t)NN);
  k_zero<<<CDIV(NN, T), T, 0, stream>>>(gAx, (size_t)NN);
  k_spmm_s<<<CDIV(EE, T), T, 0, stream>>>(erow, ecol, wEd, g1, gAx);
  k_jacobi<<<CDIV(NN, T), T, 0, stream>>>(coefs + 8, g1, g0, gAx, g0, gO, 0, (size_t)NN);
  k_zero<<<CDIV(NN, T), T, 0, stream>>>(gAx, (size_t)NN);
  k_spmm_s<<<CDIV(EE, T), T, 0, stream>>>(erow, ecol, wEd, g0, gAx);
  k_jacobi<<<CDIV(NN, T), T, 0, stream>>>(coefs + 12, g0, g1, gAx, g1, gO, 0, (size_t)NN);
  // gO == g  (t1 = g⊗b2, t2 = g⊗b3)

  // 5) Tucker head
  k_headP0<<<FF, 256, 0, stream>>>(bufO, U0, P0);
  k_headH<<<1, 256, 0, stream>>>(gO, U0, hbuf);
  k_final<<<1, 64, 0, stream>>>(P0, hbuf, b2, b3, U1, U2, core, U3, trlb, outp);
}